// MultiHeadAttentionBlock_1889785610644
// MI455X (gfx1250) — compile-verified
//
#include <hip/hip_runtime.h>

typedef __attribute__((ext_vector_type(2))) float v2f;
typedef __attribute__((ext_vector_type(8))) float v8f;

// B=2, S=2048, D=1024, H=16, DK=64. Rows flattened: M = B*S = 4096.
#define DIM_D 1024
#define ROWS  4096
#define SEQ   2048
#define HEADS 16
#define DK    64

// ---------------------------------------------------------------------------
// C[M,N] = A[M,K] @ W[N,K]^T + bias[N]       (torch Linear: W is [out,in])
// Tile: 128x128 per workgroup, 8 waves (2 M x 4 N), wave tile 64x32.
// WMMA f32 16x16x4: A-frag & B-frag both read float2 at
//   row = tileRow + (lane&15), col = k0 + (lane>=16 ? 2 : 0)
// ---------------------------------------------------------------------------
__global__ __launch_bounds__(256) void gemm_bias_wmma(
    const float* __restrict__ A, const float* __restrict__ W,
    const float* __restrict__ bias, float* __restrict__ C,
    int M, int N, int K)
{
  __shared__ float la[128][20];   // stride 20 floats: conflict-free b64 frag reads
  __shared__ float lw[128][20];

  const int tid  = threadIdx.x;
  const int lane = tid & 31;
  const int wave = tid >> 5;
  const int wm   = (wave >> 2) * 64;   // 0 / 64
  const int wn   = (wave & 3) * 32;    // 0,32,64,96

  const int m0 = blockIdx.y * 128;
  const int n0 = blockIdx.x * 128;

  const int rl    = lane & 15;
  const int half2 = (lane >> 4) << 1;  // 0 or 2

  v8f zero = {};
  v8f acc[4][2];
#pragma unroll
  for (int i = 0; i < 4; i++)
#pragma unroll
    for (int j = 0; j < 2; j++) acc[i][j] = zero;

  for (int k0 = 0; k0 < K; k0 += 16) {
    // Stage 128x16 of A and of W (512 float4 each; 2 per thread per matrix).
#pragma unroll
    for (int i = 0; i < 2; i++) {
      int id = tid + i * 256;
      int r  = id >> 2;
      int c  = (id & 3) << 2;
      *(float4*)&la[r][c] = *(const float4*)&A[(size_t)(m0 + r) * K + k0 + c];
      *(float4*)&lw[r][c] = *(const float4*)&W[(size_t)(n0 + r) * K + k0 + c];
    }
    if (k0 + 16 < K) {   // pull next K-tile toward the caches
      __builtin_prefetch(&A[(size_t)(m0 + (tid & 127)) * K + k0 + 16], 0, 1);
      __builtin_prefetch(&W[(size_t)(n0 + (tid & 127)) * K + k0 + 16], 0, 1);
    }
    __syncthreads();

#pragma unroll
    for (int ks = 0; ks < 16; ks += 4) {
      v2f af[4], bf[2];
#pragma unroll
      for (int i = 0; i < 4; i++)
        af[i] = *(const v2f*)&la[wm + i * 16 + rl][ks + half2];
#pragma unroll
      for (int j = 0; j < 2; j++)
        bf[j] = *(const v2f*)&lw[wn + j * 16 + rl][ks + half2];
#pragma unroll
      for (int i = 0; i < 4; i++)
#pragma unroll
        for (int j = 0; j < 2; j++)
          acc[i][j] = __builtin_amdgcn_wmma_f32_16x16x4_f32(
              false, af[i], false, bf[j], (short)0, acc[i][j], false, false);
    }
    __syncthreads();
  }

  // C/D layout: VGPR v -> row v (lanes 0-15) / row v+8 (lanes 16-31), col = lane&15
  const int hi8 = (lane >> 4) * 8;
#pragma unroll
  for (int j = 0; j < 2; j++) {
    int col  = n0 + wn + j * 16 + rl;
    float bv = bias[col];
#pragma unroll
    for (int i = 0; i < 4; i++) {
      int rbase = m0 + wm + i * 16 + hi8;
#pragma unroll
      for (int v = 0; v < 8; v++)
        C[(size_t)(rbase + v) * N + col] = acc[i][j][v] + bv;
    }
  }
}

// ---------------------------------------------------------------------------
// M[b,h] = Kproj[b,:,h*64:+64]^T @ Vproj[b,:,h*64:+64] / sqrt(DK)   (64x64)
// One workgroup per (b,h); LDS-staged outer-product accumulation.
// ---------------------------------------------------------------------------
__global__ __launch_bounds__(256) void ktv_kernel(
    const float* __restrict__ Kp, const float* __restrict__ Vp,
    float* __restrict__ Mout)
{
  __shared__ float lk[64][64];
  __shared__ float lv[64][64];

  const int bh = blockIdx.x;            // b*16 + h
  const int b  = bh >> 4, h = bh & 15;
  const float* kbase = Kp + (size_t)b * SEQ * DIM_D + h * DK;
  const float* vbase = Vp + (size_t)b * SEQ * DIM_D + h * DK;

  const int tid = threadIdx.x;
  const int ti  = tid >> 4;             // output row group (x4)
  const int tj  = tid & 15;             // output col group (x4)

  float acc[4][4] = {};

  for (int s0 = 0; s0 < SEQ; s0 += 64) {
#pragma unroll
    for (int i = 0; i < 4; i++) {
      int id = tid + i * 256;           // 0..1023
      int r  = id >> 4;                 // 0..63
      int c  = (id & 15) << 2;          // 0..60
      *(float4*)&lk[r][c] = *(const float4*)&kbase[(size_t)(s0 + r) * DIM_D + c];
      *(float4*)&lv[r][c] = *(const float4*)&vbase[(size_t)(s0 + r) * DIM_D + c];
    }
    __syncthreads();
    for (int ss = 0; ss < 64; ss++) {
      float4 kq = *(const float4*)&lk[ss][ti * 4];
      float4 vq = *(const float4*)&lv[ss][tj * 4];
      float kr[4] = {kq.x, kq.y, kq.z, kq.w};
      float vr[4] = {vq.x, vq.y, vq.z, vq.w};
#pragma unroll
      for (int i = 0; i < 4; i++)
#pragma unroll
        for (int j = 0; j < 4; j++) acc[i][j] += kr[i] * vr[j];
    }
    __syncthreads();
  }

  float* mo = Mout + (size_t)bh * DK * DK;
#pragma unroll
  for (int i = 0; i < 4; i++)
#pragma unroll
    for (int j = 0; j < 4; j++)
      mo[(ti * 4 + i) * DK + (tj * 4 + j)] = acc[i][j] * 0.125f;  // 1/sqrt(64)
}

// ---------------------------------------------------------------------------
// attn[:, h*64:+64] = Qproj[:, h*64:+64] @ M[b,h]      (K = 64, WMMA)
// grid = (H, ROWS/128); 8 waves x 16 rows; M_h staged transposed into LDS.
// ---------------------------------------------------------------------------
__global__ __launch_bounds__(256) void qm_wmma(
    const float* __restrict__ Qp, const float* __restrict__ Mh,
    float* __restrict__ At)
{
  __shared__ float lmt[64][68];         // lmt[n][k] = M_h[k][n]

  const int h  = blockIdx.x;
  const int r0 = blockIdx.y * 128;
  const int b  = r0 >> 11;              // row / 2048
  const float* mbase = Mh + (size_t)(b * HEADS + h) * DK * DK;

  const int tid = threadIdx.x;
#pragma unroll
  for (int i = 0; i < 16; i++) {
    int id = tid + i * 256;             // 0..4095, id = k*64 + n
    lmt[id & 63][id >> 6] = mbase[id];
  }
  __syncthreads();

  const int lane  = tid & 31;
  const int wave  = tid >> 5;
  const int wr    = r0 + wave * 16;
  const int rl    = lane & 15;
  const int half2 = (lane >> 4) << 1;

  v8f zero = {};
  v8f acc[4] = {zero, zero, zero, zero};

#pragma unroll
  for (int k0 = 0; k0 < DK; k0 += 4) {
    v2f a = *(const v2f*)&Qp[(size_t)(wr + rl) * DIM_D + h * DK + k0 + half2];
#pragma unroll
    for (int j = 0; j < 4; j++) {
      v2f bf = *(const v2f*)&lmt[j * 16 + rl][k0 + half2];
      acc[j] = __builtin_amdgcn_wmma_f32_16x16x4_f32(
          false, a, false, bf, (short)0, acc[j], false, false);
    }
  }

  const int hi8 = (lane >> 4) * 8;
#pragma unroll
  for (int j = 0; j < 4; j++) {
    int col = h * DK + j * 16 + rl;
#pragma unroll
    for (int v = 0; v < 8; v++)
      At[(size_t)(wr + v + hi8) * DIM_D + col] = acc[j][v];
  }
}

// ---------------------------------------------------------------------------
extern "C" void kernel_launch(void* const* d_in, const int* in_sizes, int n_in,
                              void* d_out, int out_size, void* d_ws, size_t ws_size,
                              hipStream_t stream) {
  const float* q   = (const float*)d_in[0];
  const float* k   = (const float*)d_in[1];
  const float* v   = (const float*)d_in[2];
  const float* w_q = (const float*)d_in[3];
  const float* b_q = (const float*)d_in[4];
  const float* w_k = (const float*)d_in[5];
  const float* b_k = (const float*)d_in[6];
  const float* w_v = (const float*)d_in[7];
  const float* b_v = (const float*)d_in[8];
  const float* w_o = (const float*)d_in[9];
  const float* b_o = (const float*)d_in[10];
  float* out = (float*)d_out;

  float* ws = (float*)d_ws;
  const size_t SZ = (size_t)ROWS * DIM_D;       // 4096*1024
  float* Qp = ws;
  float* Kp = ws + SZ;
  float* Vp = ws + 2 * SZ;
  float* Mb = ws + 3 * SZ;                      // 32 * 64 * 64
  float* At = Mb + (size_t)2 * HEADS * DK * DK;

  dim3 gg(DIM_D / 128, ROWS / 128);             // 8 x 32
  dim3 gb(256);

  gemm_bias_wmma<<<gg, gb, 0, stream>>>(q, w_q, b_q, Qp, ROWS, DIM_D, DIM_D);
  gemm_bias_wmma<<<gg, gb, 0, stream>>>(k, w_k, b_k, Kp, ROWS, DIM_D, DIM_D);
  gemm_bias_wmma<<<gg, gb, 0, stream>>>(v, w_v, b_v, Vp, ROWS, DIM_D, DIM_D);

  ktv_kernel<<<dim3(2 * HEADS), gb, 0, stream>>>(Kp, Vp, Mb);

  qm_wmma<<<dim3(HEADS, ROWS / 128), gb, 0, stream>>>(Qp, Mb, At);

  gemm_bias_wmma<<<gg, gb, 0, stream>>>(At, w_o, b_o, out, ROWS, DIM_D, DIM_D);
}